// GraphEdgeConvolution_78692390797706
// MI455X (gfx1250) — compile-verified
//
#include <hip/hip_runtime.h>
#include <hip/hip_bf16.h>

// Problem constants (match reference)
#define N_NODES 50000
#define N_EDGES 800000
#define D_IN    512
#define UNITS   256

typedef __attribute__((ext_vector_type(2))) float v2f;
typedef __attribute__((ext_vector_type(8))) float v8f;
typedef int v4i __attribute__((vector_size(16)));          // matches builtin's pointee
typedef __attribute__((address_space(1))) v4i glob_v4i;    // global (device) AS
typedef __attribute__((address_space(3))) v4i lds_v4i;     // LDS AS (32-bit ptr)

#if __has_builtin(__builtin_amdgcn_global_load_async_to_lds_b128) && \
    __has_builtin(__builtin_amdgcn_s_wait_asynccnt)
#define USE_ASYNC_LDS 1
#else
#define USE_ASYNC_LDS 0
#endif

// 16-byte global -> LDS copy.
__device__ __forceinline__ void copy16_to_lds(const float* gsrc, float* ldst) {
#if USE_ASYNC_LDS
    // Async DMA into LDS, tracked on ASYNCcnt (no VGPR round trip).
    // LDS flat address: low 32 bits are the LDS byte offset (ISA aperture rule).
    __builtin_amdgcn_global_load_async_to_lds_b128(
        (glob_v4i*)(uintptr_t)gsrc,
        (lds_v4i*)(uint32_t)(uintptr_t)ldst,
        0, 0);
#else
    *(float4*)ldst = *(const float4*)gsrc;
#endif
}

__device__ __forceinline__ void wait_lds_copies() {
#if USE_ASYNC_LDS
    __builtin_amdgcn_s_wait_asynccnt(0);
#endif
}

// ---------------------------------------------------------------------------
// Kernel 1: zero the output accumulator (harness poisons d_out with 0xAA)
// ---------------------------------------------------------------------------
__global__ __launch_bounds__(256) void zero_out_kernel(float4* out4, int n4) {
    int i = blockIdx.x * 256 + threadIdx.x;
    if (i < n4) out4[i] = make_float4(0.f, 0.f, 0.f, 0.f);
}

// ---------------------------------------------------------------------------
// Kernel 2: h = x @ kernel  via V_WMMA_F32_16X16X4_F32
//   block = 256 threads (8 waves). Block tile: 128 (M) x 64 (N).
//   Wave w owns rows [blockM + 16w, +16) and 4 adjacent 16-col C tiles
//   (4 accumulators -> each A fragment feeds 4 WMMAs).
//   K staged through LDS in chunks of 64: 16 K-steps x 4 WMMA per chunk.
// LDS: As[128][68]  (stride 68 -> A-frag reads hit 16 distinct banks/half),
//      Bs[64][80]   (stride 80 -> the two half-wave B reads use disjoint banks)
// ---------------------------------------------------------------------------
#define KC     64
#define AS_LD  68
#define BS_LD  80

__global__ __launch_bounds__(256) void gemm_wmma_kernel(const float* __restrict__ x,
                                                        const float* __restrict__ w,
                                                        float* __restrict__ h) {
    __shared__ float As[128 * AS_LD];   // 34816 B
    __shared__ float Bs[KC * BS_LD];    // 20480 B

    const int tid    = threadIdx.x;
    const int lane   = tid & 31;
    const int wave   = tid >> 5;          // 0..7
    const int blockM = blockIdx.x * 128;
    const int nbase  = blockIdx.y * 64;

    const int m      = lane & 15;         // fragment row/col within 16
    const int khalf  = (lane >> 4) * 2;   // 0 or 2 (K offset per ISA layout)

    v8f acc[4] = {};

    for (int kb = 0; kb < D_IN; kb += KC) {
        // ---- stage A: 128 rows x 64 cols = 2048 float4, 8 per thread ----
        #pragma unroll
        for (int i = 0; i < 8; ++i) {
            int idx  = tid + i * 256;
            int row  = idx >> 4;          // 0..127
            int c4   = idx & 15;          // float4 index within 64-col chunk
            int gRow = blockM + row;
            if (gRow >= N_NODES) gRow = N_NODES - 1;   // clamp; stores guarded
            copy16_to_lds(x + (size_t)gRow * D_IN + kb + c4 * 4,
                          &As[row * AS_LD + c4 * 4]);
        }
        // ---- stage B: 64 rows x 64 cols = 1024 float4, 4 per thread ----
        #pragma unroll
        for (int i = 0; i < 4; ++i) {
            int idx = tid + i * 256;
            int row = idx >> 4;           // 0..63
            int c4  = idx & 15;           // float4 index within 64-col chunk
            copy16_to_lds(w + (size_t)(kb + row) * UNITS + nbase + c4 * 4,
                          &Bs[row * BS_LD + c4 * 4]);
        }
        wait_lds_copies();
        __syncthreads();

        // ---- compute: 16 K-steps (K=4), 4 N-tiles each ----
        const int wm = wave * 16;
        #pragma unroll
        for (int kk = 0; kk < KC; kk += 4) {
            v2f a;
            // A fragment (ISA 7.12.2): lane<16 -> K=kk,kk+1 ; lane>=16 -> kk+2,kk+3
            a.x = As[(wm + m) * AS_LD + kk + khalf + 0];
            a.y = As[(wm + m) * AS_LD + kk + khalf + 1];
            #pragma unroll
            for (int j = 0; j < 4; ++j) {
                v2f b;
                b.x = Bs[(kk + khalf + 0) * BS_LD + j * 16 + m];
                b.y = Bs[(kk + khalf + 1) * BS_LD + j * 16 + m];
                // 8 args: (neg_a, A, neg_b, B, c_mod, C, reuse_a, reuse_b)
                acc[j] = __builtin_amdgcn_wmma_f32_16x16x4_f32(false, a, false, b,
                                                               (short)0, acc[j],
                                                               false, false);
            }
        }
        __syncthreads();
    }

    // ---- store C tiles (ISA 7.12.2, 32-bit C/D 16x16 layout) ----
    const int rowBase = blockM + wave * 16 + (lane >> 4) * 8;
    #pragma unroll
    for (int j = 0; j < 4; ++j) {
        const int col = nbase + j * 16 + m;
        #pragma unroll
        for (int r = 0; r < 8; ++r) {
            int row = rowBase + r;
            if (row < N_NODES) h[(size_t)row * UNITS + col] = acc[j][r];
        }
    }
}

// ---------------------------------------------------------------------------
// Kernel 3: message passing. 4 edges per block; 64 lanes/edge, float4 each
// (64 * 4 = 256 channels). Gather h[col] (h = 51 MB -> L2-resident on the
// 192 MB L2), scale by edge weight, scatter-add with non-returning f32
// atomics (global_atomic_add_f32, STOREcnt-tracked).
// ---------------------------------------------------------------------------
__global__ __launch_bounds__(256) void scatter_kernel(const float4* __restrict__ h4,
                                                      const int* __restrict__ edge_index,
                                                      const float* __restrict__ edge_weight,
                                                      float* __restrict__ out) {
    const int e  = blockIdx.x * 4 + (threadIdx.x >> 6);   // edge id
    const int c4 = threadIdx.x & 63;                      // float4 lane within row
    if (e >= N_EDGES) return;

    const int   row = edge_index[2 * e + 0];
    const int   col = edge_index[2 * e + 1];
    const float wgt = edge_weight[e];

    float4 v = h4[(size_t)col * (UNITS / 4) + c4];
    v.x *= wgt; v.y *= wgt; v.z *= wgt; v.w *= wgt;

    float* dst = out + (size_t)row * UNITS + c4 * 4;
    unsafeAtomicAdd(dst + 0, v.x);
    unsafeAtomicAdd(dst + 1, v.y);
    unsafeAtomicAdd(dst + 2, v.z);
    unsafeAtomicAdd(dst + 3, v.w);
}

// ---------------------------------------------------------------------------
extern "C" void kernel_launch(void* const* d_in, const int* in_sizes, int n_in,
                              void* d_out, int out_size, void* d_ws, size_t ws_size,
                              hipStream_t stream) {
    const float* x  = (const float*)d_in[0];
    const int*   ei = (const int*)d_in[1];
    const float* ew = (const float*)d_in[2];
    const float* wk = (const float*)d_in[3];
    float* out = (float*)d_out;
    float* h   = (float*)d_ws;      // 50000*256*4 = 51.2 MB scratch

    // 1) zero output
    {
        int n4 = (N_NODES * UNITS) / 4;                 // 3,200,000
        int blocks = (n4 + 255) / 256;
        zero_out_kernel<<<blocks, 256, 0, stream>>>((float4*)out, n4);
    }
    // 2) GEMM h = x @ kernel
    {
        dim3 grid((N_NODES + 127) / 128, UNITS / 64);   // 391 x 4
        gemm_wmma_kernel<<<grid, 256, 0, stream>>>(x, wk, h);
    }
    // 3) gather-scale-scatter over edges
    {
        int blocks = N_EDGES / 4;                       // 200,000
        scatter_kernel<<<blocks, 256, 0, stream>>>((const float4*)h, ei, ew, out);
    }
}